// SPDBatchNormMeanVar_22643067584714
// MI455X (gfx1250) — compile-verified
//
#include <hip/hip_runtime.h>
#include <math.h>

// ---------------------------------------------------------------------------
// SPD Riemannian batch-norm for MI455X (gfx1250, wave32, WMMA f32 16x16x4)
// B=8192 SPD matrices, n=64.  All GEMMs run on v_wmma_f32_16x16x4_f32.
// Eigendecompositions: batched parallel cyclic Jacobi in LDS.
// ---------------------------------------------------------------------------

#define PITCH 65           // LDS row pitch (floats) -> no 64-bank conflicts
#define BATCH 8192
#define MATSZ 4096         // 64*64
#define SWEEPS 10          // Jacobi sweeps (63 rounds each)

typedef __attribute__((ext_vector_type(2))) float v2f;
typedef __attribute__((ext_vector_type(8))) float v8f;

// workspace layout (float offsets)
#define WS_MEAN  ((size_t)0)
#define WS_GS    ((size_t)4096)
#define WS_GIS   ((size_t)8192)
#define WS_LSUM  ((size_t)12288)
#define WS_GIS2  ((size_t)16384)
#define WS_BS    ((size_t)20480)
#define WS_SCAL  ((size_t)24576)
#define WS_PART  ((size_t)24640)                      // 64*4096 partials
#define WS_LOGW  (WS_PART + (size_t)64 * 4096)        // 8192*64
#define WS_BIG   (WS_LOGW + (size_t)BATCH * 64)       // 8192*4096 staging

// ---------------------------------------------------------------------------
// 64x64x64 fp32 GEMM via V_WMMA_F32_16X16X4_F32.
// 8 waves/block, each wave computes two 16x16 C tiles.
// A-fragment (16x4): lane L -> M = L%16, regs j: K = k0 + j + 2*(L/16)
// B-fragment (4x16): mirrored: N = L%16, K = k0 + j + 2*(L/16)
// C/D: reg r, lane L -> (M = tm + r + 8*(L/16), N = tn + L%16)
// ---------------------------------------------------------------------------
__device__ __forceinline__ void gemm64(const float* A, int lda,
                                       const float* B, int ldb, bool transB,
                                       float* C, int ldc)
{
  const int lane = threadIdx.x & 31;
  const int wave = threadIdx.x >> 5;
  const int half = lane >> 4;
  const int l16  = lane & 15;
  for (int t = wave; t < 16; t += 8) {
    const int tm = (t >> 2) << 4;
    const int tn = (t & 3) << 4;
    v8f acc = {0.f, 0.f, 0.f, 0.f, 0.f, 0.f, 0.f, 0.f};
#pragma unroll
    for (int k0 = 0; k0 < 64; k0 += 4) {
      const int ka = k0 + (half << 1);
      v2f av, bv;
      av.x = A[(tm + l16) * lda + ka];
      av.y = A[(tm + l16) * lda + ka + 1];
      if (!transB) {
        bv.x = B[ka * ldb + tn + l16];
        bv.y = B[(ka + 1) * ldb + tn + l16];
      } else {
        bv.x = B[(tn + l16) * ldb + ka];
        bv.y = B[(tn + l16) * ldb + ka + 1];
      }
      acc = __builtin_amdgcn_wmma_f32_16x16x4_f32(
          false, av, false, bv, (short)0, acc, false, false);
    }
#pragma unroll
    for (int r = 0; r < 8; ++r)
      C[(size_t)(tm + r + (half << 3)) * ldc + tn + l16] = acc[r];
  }
}

// global (row-major 64) -> LDS (pitch 65), float4 loads
__device__ __forceinline__ void load64(const float* __restrict__ g, float* l)
{
  const float4* g4 = (const float4*)g;
  for (int t = threadIdx.x; t < 1024; t += 256) {
    float4 v = g4[t];
    const int i = t >> 4, j = (t & 15) << 2;
    float* d = &l[i * PITCH + j];
    d[0] = v.x; d[1] = v.y; d[2] = v.z; d[3] = v.w;
  }
}

__device__ __forceinline__ void store64(const float* l, float* __restrict__ g)
{
  float4* g4 = (float4*)g;
  for (int t = threadIdx.x; t < 1024; t += 256) {
    const int i = t >> 4, j = (t & 15) << 2;
    const float* s = &l[i * PITCH + j];
    float4 v; v.x = s[0]; v.y = s[1]; v.z = s[2]; v.w = s[3];
    g4[t] = v;
  }
}

// ---------------------------------------------------------------------------
// Parallel cyclic Jacobi eigensolver, symmetric 64x64 in LDS.
// A (pitch 65) destroyed; V <- eigenvectors (columns), w <- eigenvalues.
// Eigenvalue order is irrelevant: all uses are V f(w) V^T (order-invariant).
// ---------------------------------------------------------------------------
__device__ void jacobi64(float* A, float* V, float* w)
{
  __shared__ float jc[32], js[32];
  __shared__ int jp[32], jq[32], pos[64];
  const int tid = threadIdx.x;

  for (int t = tid; t < 4096; t += 256) {
    const int i = t >> 6, j = t & 63;
    V[i * PITCH + j] = (i == j) ? 1.0f : 0.0f;
  }
  if (tid < 64) pos[tid] = tid;
  __syncthreads();

  for (int sweep = 0; sweep < SWEEPS; ++sweep) {
    for (int round = 0; round < 63; ++round) {
      if (tid < 32) {
        int p = pos[tid], q = pos[63 - tid];
        if (p > q) { int tmp = p; p = q; q = tmp; }
        jp[tid] = p; jq[tid] = q;
        const float app = A[p * PITCH + p];
        const float aqq = A[q * PITCH + q];
        const float apq = A[p * PITCH + q];
        float c = 1.0f, s = 0.0f;
        if (fabsf(apq) > 1e-30f) {
          const float tau = (aqq - app) / (2.0f * apq);
          const float tt = (tau >= 0.0f ? 1.0f : -1.0f) /
                           (fabsf(tau) + sqrtf(1.0f + tau * tau));
          c = 1.0f / sqrtf(1.0f + tt * tt);
          s = tt * c;
        }
        jc[tid] = c; js[tid] = s;
      }
      __syncthreads();
      // right rotation A <- A J, V <- V J  (disjoint column pairs)
      for (int job = tid; job < 2048; job += 256) {
        const int k = job >> 6, i = job & 63;
        const int p = jp[k], q = jq[k];
        const float c = jc[k], s = js[k];
        const float aip = A[i * PITCH + p], aiq = A[i * PITCH + q];
        A[i * PITCH + p] = c * aip - s * aiq;
        A[i * PITCH + q] = s * aip + c * aiq;
        const float vip = V[i * PITCH + p], viq = V[i * PITCH + q];
        V[i * PITCH + p] = c * vip - s * viq;
        V[i * PITCH + q] = s * vip + c * viq;
      }
      __syncthreads();
      // left rotation A <- J^T A  (disjoint row pairs)
      for (int job = tid; job < 2048; job += 256) {
        const int k = job >> 6, j = job & 63;
        const int p = jp[k], q = jq[k];
        const float c = jc[k], s = js[k];
        const float apj = A[p * PITCH + j], aqj = A[q * PITCH + j];
        A[p * PITCH + j] = c * apj - s * aqj;
        A[q * PITCH + j] = s * apj + c * aqj;
      }
      __syncthreads();
      if (tid == 0) {   // tournament rotation, pos[0] fixed
        const int last = pos[63];
        for (int i = 63; i > 1; --i) pos[i] = pos[i - 1];
        pos[1] = last;
      }
      __syncthreads();
    }
  }
  if (tid < 64) w[tid] = A[tid * PITCH + tid];
  __syncthreads();
}

// f(w) modes
__device__ __forceinline__ float apply_f(float x, int mode, float s)
{
  switch (mode) {
    case 0:  return sqrtf(x);
    case 1:  return 1.0f / sqrtf(x);
    case 2:  return logf(x);
    case 3:  return expf(x);
    default: return expf(s * x);   // x is log(eigenvalue) here
  }
}

// out = V diag(f(w)) V^T  (WMMA; Wscratch is an LDS matrix)
__device__ void recon(const float* V, const float* w, int mode, float s,
                      float* Wscratch, float* out, int ldo)
{
  __shared__ float fw[64];
  const int tid = threadIdx.x;
  if (tid < 64) fw[tid] = apply_f(w[tid], mode, s);
  __syncthreads();
  for (int t = tid; t < 4096; t += 256) {
    const int i = t >> 6, j = t & 63;
    Wscratch[i * PITCH + j] = V[i * PITCH + j] * fw[j];
  }
  __syncthreads();
  gemm64(Wscratch, PITCH, V, PITCH, true, out, ldo);
  __syncthreads();
}

// ---------------------------------------------------------------------------
// Kernels
// ---------------------------------------------------------------------------

// partial sums over the batch axis: part[by*4096 + t] = sum of 128 matrices
__global__ void k_bigreduce(const float* __restrict__ src, float* __restrict__ part)
{
  const int t = blockIdx.x * 256 + threadIdx.x;          // 0..4095
  const int b0 = blockIdx.y * 128;
  float sum = 0.0f;
  for (int b = 0; b < 128; ++b)
    sum += src[(size_t)(b0 + b) * MATSZ + t];
  part[(size_t)blockIdx.y * MATSZ + t] = sum;
}

// dst[t] = scale * sum_{64 partials}
__global__ void k_part2(const float* __restrict__ part, float* __restrict__ dst,
                        float scale)
{
  const int t = blockIdx.x * 256 + threadIdx.x;
  float sum = 0.0f;
  for (int p = 0; p < 64; ++p)
    sum += part[(size_t)p * MATSZ + t];
  dst[t] = sum * scale;
}

// block 0: mean -> Gs, Gis ; block 1: bias -> Bs
__global__ void k_prep(const float* __restrict__ bias, float* __restrict__ ws)
{
  __shared__ float MA[64 * PITCH], MV[64 * PITCH];
  __shared__ float w[64];
  if (blockIdx.x == 0) {
    load64(ws + WS_MEAN, MA);
    __syncthreads();
    jacobi64(MA, MV, w);                   // MA destroyed -> reuse as scratch
    recon(MV, w, 0, 0.f, MA, ws + WS_GS, 64);
    recon(MV, w, 1, 0.f, MA, ws + WS_GIS, 64);
  } else {
    load64(bias, MA);
    __syncthreads();
    jacobi64(MA, MV, w);
    recon(MV, w, 0, 0.f, MA, ws + WS_BS, 64);
  }
}

// per batch: Lb = logm(Gis x Gis) -> BIG staging
__global__ void k_karcher(const float* __restrict__ x, float* __restrict__ ws)
{
  __shared__ float MA[64 * PITCH], MB[64 * PITCH], MC[64 * PITCH];
  __shared__ float w[64];
  const size_t b = blockIdx.x;
  load64(ws + WS_GIS, MA);
  load64(x + b * MATSZ, MB);
  __syncthreads();
  gemm64(MA, PITCH, MB, PITCH, false, MC, PITCH);   // MC = Gis @ x
  __syncthreads();
  gemm64(MC, PITCH, MA, PITCH, false, MB, PITCH);   // MB = Gis @ x @ Gis
  __syncthreads();
  jacobi64(MB, MC, w);                              // MC = V
  recon(MC, w, 2, 0.f, MA, ws + WS_BIG + b * MATSZ, 64);
}

// single block: L -> expm -> mean2 = Gs E Gs -> Gis2 = mean2^{-1/2}
__global__ void k_update(float* __restrict__ ws)
{
  __shared__ float MA[64 * PITCH], MB[64 * PITCH], MC[64 * PITCH];
  __shared__ float w[64];
  load64(ws + WS_LSUM, MA);
  __syncthreads();
  jacobi64(MA, MB, w);                              // eig(L), MB = V
  recon(MB, w, 3, 0.f, MA, MC, PITCH);              // MC = expm(L)
  load64(ws + WS_GS, MA);
  __syncthreads();
  gemm64(MA, PITCH, MC, PITCH, false, MB, PITCH);   // MB = Gs @ E
  __syncthreads();
  gemm64(MB, PITCH, MA, PITCH, false, MC, PITCH);   // MC = mean2
  __syncthreads();
  jacobi64(MC, MB, w);                              // eig(mean2), MB = V
  recon(MB, w, 1, 0.f, MA, ws + WS_GIS2, 64);       // Gis2
}

// per batch: centered = Gis2 x Gis2; eig -> store V (BIG) and log(w) (LOGW)
__global__ void k_center(const float* __restrict__ x, float* __restrict__ ws)
{
  __shared__ float MA[64 * PITCH], MB[64 * PITCH], MC[64 * PITCH];
  __shared__ float w[64];
  const size_t b = blockIdx.x;
  load64(ws + WS_GIS2, MA);
  load64(x + b * MATSZ, MB);
  __syncthreads();
  gemm64(MA, PITCH, MB, PITCH, false, MC, PITCH);
  __syncthreads();
  gemm64(MC, PITCH, MA, PITCH, false, MB, PITCH);   // MB = centered
  __syncthreads();
  jacobi64(MB, MC, w);                              // MC = V
  if (threadIdx.x < 64)
    ws[WS_LOGW + b * 64 + threadIdx.x] = logf(w[threadIdx.x]);
  store64(MC, ws + WS_BIG + b * MATSZ);
}

// single block deterministic reduction: var = mean_b sum_j logw^2 ; s = w/sqrt(var+eps)
__global__ void k_var(const float* __restrict__ weight, float* __restrict__ ws)
{
  __shared__ float red[256];
  const int tid = threadIdx.x;
  float sum = 0.0f;
  for (size_t i = tid; i < (size_t)BATCH * 64; i += 256) {
    const float v = ws[WS_LOGW + i];
    sum += v * v;
  }
  red[tid] = sum;
  __syncthreads();
  for (int s = 128; s > 0; s >>= 1) {
    if (tid < s) red[tid] += red[tid + s];
    __syncthreads();
  }
  if (tid == 0) {
    const float var = red[0] / (float)BATCH;
    ws[WS_SCAL] = weight[0] / sqrtf(var + 1e-5f);
  }
}

// per batch: out = Bs (V diag(exp(s*logw)) V^T) Bs
__global__ void k_final(float* __restrict__ ws, float* __restrict__ out)
{
  __shared__ float MA[64 * PITCH], MB[64 * PITCH], MC[64 * PITCH];
  __shared__ float lw[64];
  const size_t b = blockIdx.x;
  load64(ws + WS_BIG + b * MATSZ, MC);              // V
  if (threadIdx.x < 64)
    lw[threadIdx.x] = ws[WS_LOGW + b * 64 + threadIdx.x];
  const float s = ws[WS_SCAL];
  __syncthreads();
  recon(MC, lw, 4, s, MA, MB, PITCH);               // MB = normalized
  load64(ws + WS_BS, MA);
  __syncthreads();
  gemm64(MA, PITCH, MB, PITCH, false, MC, PITCH);   // MC = Bs @ N
  __syncthreads();
  gemm64(MC, PITCH, MA, PITCH, false, out + b * MATSZ, 64);
}

// ---------------------------------------------------------------------------
extern "C" void kernel_launch(void* const* d_in, const int* in_sizes, int n_in,
                              void* d_out, int out_size, void* d_ws, size_t ws_size,
                              hipStream_t stream)
{
  const float* x      = (const float*)d_in[0];   // (8192, 64, 64) f32
  const float* bias   = (const float*)d_in[1];   // (64, 64) f32
  const float* weight = (const float*)d_in[2];   // scalar f32
  float* out = (float*)d_out;
  float* ws  = (float*)d_ws;

  const dim3 blk(256);
  const float invB = 1.0f / (float)BATCH;

  // 1) batch mean (two-stage deterministic reduction)
  k_bigreduce<<<dim3(16, 64), blk, 0, stream>>>(x, ws + WS_PART);
  k_part2<<<16, blk, 0, stream>>>(ws + WS_PART, ws + WS_MEAN, invB);
  // 2) eig(mean) -> Gs, Gis ; eig(bias) -> Bs
  k_prep<<<2, blk, 0, stream>>>(bias, ws);
  // 3) Karcher step tangent logs
  k_karcher<<<BATCH, blk, 0, stream>>>(x, ws);
  // 4) L = mean of logs
  k_bigreduce<<<dim3(16, 64), blk, 0, stream>>>(ws + WS_BIG, ws + WS_PART);
  k_part2<<<16, blk, 0, stream>>>(ws + WS_PART, ws + WS_LSUM, invB);
  // 5) mean update + Gis2
  k_update<<<1, blk, 0, stream>>>(ws);
  // 6) centering + per-batch eig (store V, logw)
  k_center<<<BATCH, blk, 0, stream>>>(x, ws);
  // 7) dispersion -> scale s
  k_var<<<1, blk, 0, stream>>>(weight, ws);
  // 8) power + re-bias -> output
  k_final<<<BATCH, blk, 0, stream>>>(ws, out);
}